// Embedding2d_41901700940494
// MI455X (gfx1250) — compile-verified
//
#include <hip/hip_runtime.h>
#include <hip/hip_bf16.h>

// Embedding gather + transpose for MI455X (gfx1250, wave32).
//
//   out[b][c][h][w][t] = weight[idx[b][h][w][t]][c]
//
// Store-bandwidth-bound (128 MB out @ 23.3 TB/s ~ 5.7 us floor).
// Strategy: per-block LDS transpose tile, filled via CDNA5 async
// global->LDS DMA (GLOBAL_LOAD_ASYNC_TO_LDS_B128, ASYNCcnt), drained
// with fully coalesced contiguous stores per c-slice. Output stores are
// non-temporal so the 128 MB write stream doesn't evict the L2-resident
// 2 MB weight table.

#define NUM_EMB    8192
#define DIM        64
#define HWT        32768            // 64*64*8 positions per batch
#define NPOS       524288           // 16 * HWT
#define TILE       128              // flat positions per workgroup
#define LDS_STRIDE 68               // 64 + 4 pad (272 B rows: 16B-aligned, avoids 64-bank clash)

__global__ __launch_bounds__(256)
void Embedding2d_41901700940494_kernel(const int* __restrict__ idx,
                                       const float* __restrict__ weight,
                                       float* __restrict__ out)
{
    __shared__ int   sidx[TILE];
    __shared__ float lds[TILE * LDS_STRIDE];   // 34,816 B

    const int t       = threadIdx.x;           // 0..255 (8 waves)
    const int p0      = blockIdx.x * TILE;     // flat position base (stays inside one batch)
    const int b       = p0 >> 15;              // p0 / HWT
    const int q0      = p0 & (HWT - 1);        // p0 % HWT

    // ---- stage indices (1 coalesced load; read-once -> non-temporal) ----
    if (t < TILE) sidx[t] = __builtin_nontemporal_load(&idx[p0 + t]);
    __syncthreads();

    // LDS byte offset of tile base: generic ptr from LDS has the LDS byte
    // address in its low 32 bits (SHARED aperture lives in [63:32]).
    const unsigned ldsBase = (unsigned)(unsigned long long)(const void*)lds;

    // ---- gather: async DMA weight rows into LDS ----
    // 128 rows x 16 chunks(16B) = 2048 chunks; 8 per thread.
    // 16 consecutive lanes cover one 256 B weight row -> coalesced reads
    // (table stays hot in WGP$/L2).
#pragma unroll
    for (int j = 0; j < 8; ++j) {
        const int k   = t + 256 * j;
        const int row = k >> 4;          // 0..127
        const int sub = k & 15;          // 16B chunk within row
        const int r   = sidx[row];       // LDS broadcast read
        const char* gsrc = (const char*)weight + ((size_t)(unsigned)r << 8) + (sub << 4);
        const unsigned ldst = ldsBase + (unsigned)(row * (LDS_STRIDE * 4) + (sub << 4));
        asm volatile("global_load_async_to_lds_b128 %0, %1, off"
                     :
                     : "v"(ldst), "v"(gsrc)
                     : "memory");
    }

    // drain this wave's async DMA, then make tile visible workgroup-wide
    asm volatile("s_wait_asynccnt 0x0" ::: "memory");
    __syncthreads();

    // ---- transposed writeout: contiguous 128 B per wave per store ----
    // element e = i*256 + t ; c = 2*i + (t>>7) ; p = t & 127
    const int       tp  = t & 127;                 // position within tile
    const int       chi = t >> 7;                  // low/high c of the pair
    float*          op  = out + (size_t)b * (DIM * HWT) + (size_t)chi * HWT + q0 + tp;
    const float*    lp  = &lds[tp * LDS_STRIDE + chi];

#pragma unroll
    for (int i = 0; i < 32; ++i) {
        // c = 2i + chi; coalesced per wave; NT: pure streaming output
        __builtin_nontemporal_store(lp[2 * i], &op[(size_t)i * (2 * HWT)]);
    }
}

extern "C" void kernel_launch(void* const* d_in, const int* in_sizes, int n_in,
                              void* d_out, int out_size, void* d_ws, size_t ws_size,
                              hipStream_t stream)
{
    (void)in_sizes; (void)n_in; (void)d_ws; (void)ws_size; (void)out_size;
    const int*   idx    = (const int*)d_in[0];
    const float* weight = (const float*)d_in[1];
    float*       out    = (float*)d_out;

    const int blocks = NPOS / TILE;  // 4096
    Embedding2d_41901700940494_kernel<<<blocks, 256, 0, stream>>>(idx, weight, out);
}